// Encoder_layer_3813930959353
// MI455X (gfx1250) — compile-verified
//
#include <hip/hip_runtime.h>
#include <math.h>

// ---------------- problem constants ----------------
#define B_       2
#define SEQ_     384
#define N_AUX_   4
#define S_       1537          // SEQ*N_AUX + 1
#define DM_      512
#define NHEAD_   8
#define HD_      64
#define ROPE_    32
#define KVC_     256
#define QC_      384
#define DFF_     1536
#define H2_      16            // NHEAD*2
#define DH_      32            // HD/2
#define RH_      16            // ROPE/2
#define QKD_     48            // DH + RH
#define MROWS_   (B_*S_)       // 3074
#define EPS_     1.1920929e-07f
#define SCALING_ (1.0f/48.0f)
#define LAMINIT_ 0.2f
#define NKT_     ((S_+15)/16)  // 97 key tiles
#define SPAD_    1540          // padded LDS row length (>= S_)

typedef float v8f_t __attribute__((ext_vector_type(8)));
typedef float v2f_t __attribute__((ext_vector_type(2)));

// ---------------- helpers ----------------
__device__ __forceinline__ float wave_sum(float v) {
  #pragma unroll
  for (int off = 16; off; off >>= 1) v += __shfl_xor(v, off, 32);
  return v;
}
__device__ __forceinline__ float wave_max(float v) {
  #pragma unroll
  for (int off = 16; off; off >>= 1) v = fmaxf(v, __shfl_xor(v, off, 32));
  return v;
}
// mask: True = masked out.  col 0 never masked; row 0 attends only col 0;
// otherwise block-causal per aux tile: masked iff ((k-1)%SEQ) > ((q-1)%SEQ)
__device__ __forceinline__ bool is_masked(int q, int k) {
  if (k == 0) return false;
  if (q == 0) return true;
  return ((k - 1) % SEQ_) > ((q - 1) % SEQ_);
}

// ---------------- RMSNorm (rows) ----------------
__global__ void rmsnorm_kernel(const float* __restrict__ in, const float* __restrict__ w,
                               float* __restrict__ out, int dim, int in_stride, float eps) {
  int row = blockIdx.x;
  const float* ip = in + (size_t)row * in_stride;
  float* op = out + (size_t)row * dim;
  float ss = 0.f;
  for (int i = threadIdx.x; i < dim; i += blockDim.x) { float v = ip[i]; ss += v * v; }
  __shared__ float red[32];
  ss = wave_sum(ss);
  int wid = threadIdx.x >> 5;
  if ((threadIdx.x & 31) == 0) red[wid] = ss;
  __syncthreads();
  int nw = blockDim.x >> 5;
  if (threadIdx.x < 32) {
    float s2 = (threadIdx.x < nw) ? red[threadIdx.x] : 0.f;
    s2 = wave_sum(s2);
    if (threadIdx.x == 0) red[0] = s2;
  }
  __syncthreads();
  float scale = rsqrtf(red[0] / (float)dim + eps);
  for (int i = threadIdx.x; i < dim; i += blockDim.x) op[i] = ip[i] * scale * w[i];
}

// ---------------- fp32 WMMA GEMM: C = A(MxK) * W(KxN) [+ Res] ----------------
// one wave per 16x64 output tile (4 accumulators share one A fragment).
// K multiple of 4, N multiple of 16. Partial 64-quads handled by clamped
// loads + guarded stores (tile validity is wave-uniform: EXEC stays all-1s).
__global__ void gemm_f32_wmma(const float* __restrict__ A, const float* __restrict__ Bm,
                              const float* __restrict__ Res, float* __restrict__ C,
                              int Mrows, int K, int N) {
  int gwave = (int)((blockIdx.x * (size_t)blockDim.x + threadIdx.x) >> 5);
  int lane  = threadIdx.x & 31;
  int ntn   = N >> 4;                        // 16-col tiles
  int nquad = (ntn + 3) >> 2;                // 64-col quads
  int ntm   = (Mrows + 15) >> 4;
  if (gwave >= ntm * nquad) return;          // uniform per wave
  int tm = (gwave / nquad) * 16;
  int tq = (gwave % nquad) * 64;             // first column of quad
  int l = lane & 15;
  int khalf = (lane >> 4) * 2;               // lanes 0-15: K {0,1}; lanes 16-31: K {2,3}
  int arow = tm + l;
  float avalid = (arow < Mrows) ? 1.f : 0.f;
  if (arow >= Mrows) arow = Mrows - 1;       // clamp, zero via multiply
  const float* ap = A + (size_t)arow * K;
  // per-subtile load columns (clamped to a safe in-range address when invalid)
  int colld[4];
  #pragma unroll
  for (int t = 0; t < 4; ++t)
    colld[t] = (tq + 16 * t < N) ? (tq + 16 * t + l) : l;
  v8f_t acc[4] = {{}, {}, {}, {}};
  for (int k = 0; k < K; k += 4) {
    int kk = k + khalf;
    v2f_t a;
    a[0] = ap[kk]     * avalid;
    a[1] = ap[kk + 1] * avalid;
    const float* b0 = Bm + (size_t)kk * N;
    const float* b1 = Bm + (size_t)(kk + 1) * N;
    #pragma unroll
    for (int t = 0; t < 4; ++t) {
      v2f_t b;
      b[0] = b0[colld[t]];
      b[1] = b1[colld[t]];
      acc[t] = __builtin_amdgcn_wmma_f32_16x16x4_f32(false, a, false, b, (short)0, acc[t],
                                                     false, false);
    }
  }
  int rbase = tm + (lane >> 4) * 8;
  #pragma unroll
  for (int t = 0; t < 4; ++t) {
    int c0 = tq + 16 * t;
    if (c0 < N) {
      int col = c0 + l;
      #pragma unroll
      for (int r = 0; r < 8; ++r) {
        int row = rbase + r;
        if (row < Mrows) {
          float v = acc[t][r];
          if (Res) v += Res[(size_t)row * N + col];
          C[(size_t)row * N + col] = v;
        }
      }
    }
  }
}

// ---------------- build qh/kh (head-pair split + decoupled RoPE) ----------------
// qh/kh layout: [b][h2][s][48]  (32 nope dims + 16 rope dims)
__global__ void build_qk(const float* __restrict__ qf, const float* __restrict__ kv,
                         const float* __restrict__ ckv, const float* __restrict__ freqs,
                         float* __restrict__ qh, float* __restrict__ kh) {
  int tid = blockIdx.x * blockDim.x + threadIdx.x;
  if (tid >= B_ * S_ * H2_) return;
  int h2 = tid % H2_; int t = tid / H2_;
  int s = t % S_;     int b = t / S_;
  int n = h2 >> 1, j = h2 & 1;
  const float* qrow  = qf  + (size_t)(b * S_ + s) * (NHEAD_ * 96);
  const float* kvrow = kv  + (size_t)(b * S_ + s) * 1024;
  const float* crow  = ckv + (size_t)(b * S_ + s) * 288;
  float* qo = qh + ((size_t)(b * H2_ + h2) * S_ + s) * QKD_;
  float* ko = kh + ((size_t)(b * H2_ + h2) * S_ + s) * QKD_;
  #pragma unroll
  for (int d = 0; d < DH_; ++d) {
    qo[d] = qrow[n * 96 + j * 32 + d];
    ko[d] = kvrow[n * 128 + j * 32 + d];
  }
  float qr[RH_], kr[RH_];
  #pragma unroll
  for (int r = 0; r < RH_; ++r) {
    qr[r] = qrow[n * 96 + 64 + j * 16 + r];
    kr[r] = crow[256 + j * 16 + r];
  }
  if (s > 0) {
    int pos = (s - 1) % SEQ_;
    #pragma unroll
    for (int p = 0; p < 8; ++p) {
      float c  = freqs[((size_t)pos * 8 + p) * 2 + 0];
      float sn = freqs[((size_t)pos * 8 + p) * 2 + 1];
      float xr = qr[2 * p], xi = qr[2 * p + 1];
      qo[DH_ + 2 * p]     = xr * c - xi * sn;
      qo[DH_ + 2 * p + 1] = xr * sn + xi * c;
      xr = kr[2 * p]; xi = kr[2 * p + 1];
      ko[DH_ + 2 * p]     = xr * c - xi * sn;
      ko[DH_ + 2 * p + 1] = xr * sn + xi * c;
    }
  } else {
    #pragma unroll
    for (int r = 0; r < RH_; ++r) { qo[DH_ + r] = qr[r]; ko[DH_ + r] = kr[r]; }
  }
}

// ---------------- extract V into [b][h][s][64] ----------------
__global__ void extract_v(const float* __restrict__ kv, float* __restrict__ vT) {
  int tid = blockIdx.x * blockDim.x + threadIdx.x;
  if (tid >= B_ * S_ * NHEAD_ * HD_) return;
  int d = tid & 63; int t = tid >> 6;
  int h = t % NHEAD_; int t2 = t / NHEAD_;
  int s = t2 % S_;    int b = t2 / S_;
  vT[((size_t)(b * NHEAD_ + h) * S_ + s) * HD_ + d] =
      kv[(size_t)(b * S_ + s) * 1024 + h * 128 + 64 + d];
}

// ---------------- differential attention ----------------
// grid (NKT_, B*NHEAD), 256 threads (8 waves). LDS: p0/p1 [16][SPAD] + G[16] + lam.
__global__ void attention_kernel(const float* __restrict__ qh, const float* __restrict__ kh,
                                 const float* __restrict__ vT,
                                 const float* __restrict__ lq1, const float* __restrict__ lk1,
                                 const float* __restrict__ lq2, const float* __restrict__ lk2,
                                 float* __restrict__ o) {
  extern __shared__ float smem[];
  float* sc   = smem;                         // [2][16][SPAD_]
  float* Gv   = smem + 2 * 16 * SPAD_;        // [16]
  float* lamp = Gv + 16;                      // [1]
  int w = threadIdx.x >> 5, lane = threadIdx.x & 31;
  int bh = blockIdx.y; int b = bh / NHEAD_, h = bh % NHEAD_;
  int m0 = blockIdx.x * 16;

  // --- lambda scalar (DH_==32 == one wave) ---
  if (threadIdx.x < 32) {
    float s1 = lq1[lane] * lk1[lane];
    float s2 = lq2[lane] * lk2[lane];
    s1 = wave_sum(s1); s2 = wave_sum(s2);
    if (lane == 0) lamp[0] = expf(s1) - expf(s2) + LAMINIT_;
  }

  // --- stage 1: WMMA scores for both sub-heads into LDS ---
  // A (query) fragments depend only on (m0, j): load once, reuse across key tiles.
  int l = lane & 15;
  int khalf = (lane >> 4) * 2;
  int arow = m0 + l; if (arow >= S_) arow = S_ - 1;
  #pragma unroll
  for (int j = 0; j < 2; ++j) {
    const float* qp = qh + ((size_t)(b * H2_ + 2 * h + j) * S_ + arow) * QKD_;
    v2f_t af[QKD_ / 4];
    #pragma unroll
    for (int k = 0; k < QKD_ / 4; ++k) {
      af[k][0] = qp[4 * k + khalf];
      af[k][1] = qp[4 * k + khalf + 1];
    }
    for (int kt = w; kt < NKT_; kt += 8) {
      int n0 = kt * 16;
      int brow = n0 + l; if (brow >= S_) brow = S_ - 1;
      const float* kp = kh + ((size_t)(b * H2_ + 2 * h + j) * S_ + brow) * QKD_;
      v8f_t acc = {};
      #pragma unroll
      for (int k = 0; k < QKD_ / 4; ++k) {
        int kk = 4 * k + khalf;
        v2f_t bb;
        bb[0] = kp[kk]; bb[1] = kp[kk + 1];
        acc = __builtin_amdgcn_wmma_f32_16x16x4_f32(false, af[k], false, bb, (short)0, acc,
                                                    false, false);
      }
      int col = n0 + l;
      if (col < S_) {
        int rb = (lane >> 4) * 8;
        float* dst = sc + ((size_t)j * 16 + rb) * SPAD_ + col;
        #pragma unroll
        for (int r = 0; r < 8; ++r) dst[(size_t)r * SPAD_] = acc[r] * SCALING_;
      }
    }
  }
  __syncthreads();

  // --- stage 2: masked sign-softmax per row; G = mean(p0 row) ---
  float lam = lamp[0];
  for (int t = w; t < 32; t += 8) {
    int j = t >> 4, rr = t & 15;
    int q = m0 + rr;
    float* row = sc + ((size_t)j * 16 + rr) * SPAD_;
    float mx = -1e30f;
    for (int c = lane; c < S_; c += 32) {
      float v = row[c];
      float av = is_masked(q, c) ? -1e9f : fabsf(v);
      mx = fmaxf(mx, av);
    }
    mx = wave_max(mx);
    float sum = 0.f;
    for (int c = lane; c < S_; c += 32) {
      float v = row[c];
      float av = is_masked(q, c) ? -1e9f : fabsf(v);
      sum += expf(av - mx);
    }
    sum = wave_sum(sum);
    float inv = 1.0f / sum;
    float gs = 0.f;
    for (int c = lane; c < S_; c += 32) {
      float v = row[c];
      float av = is_masked(q, c) ? -1e9f : fabsf(v);
      float p = copysignf(expf(av - mx) * inv, v);   // masked -> exp underflows to 0
      row[c] = p;
      if (j == 0) gs += p;
    }
    if (j == 0) {
      gs = wave_sum(gs);
      if (lane == 0) Gv[rr] = gs / (float)S_;
    }
  }
  __syncthreads();

  // --- stage 3a: att = mask ? 0 : p0 - lam*p1 + G*lam  (overwrite p0) ---
  for (int t = w; t < 16; t += 8) {
    int q = m0 + t;
    float* p0 = sc + (size_t)t * SPAD_;
    const float* p1 = sc + (size_t)(16 + t) * SPAD_;
    float base = Gv[t] * lam;
    for (int c = lane; c < S_; c += 32)
      p0[c] = is_masked(q, c) ? 0.f : (p0[c] - lam * p1[c] + base);
  }
  __syncthreads();

  // --- stage 3b: o[16][64] = att @ V ---
  int d = threadIdx.x & 63;
  int rq = threadIdx.x >> 6;                   // 0..3
  float a0 = 0.f, a1 = 0.f, a2 = 0.f, a3 = 0.f;
  const float* vp = vT + (size_t)(b * NHEAD_ + h) * S_ * HD_ + d;
  for (int k = 0; k < S_; ++k) {
    float vv = vp[(size_t)k * HD_];
    a0 += sc[(size_t)(rq + 0)  * SPAD_ + k] * vv;
    a1 += sc[(size_t)(rq + 4)  * SPAD_ + k] * vv;
    a2 += sc[(size_t)(rq + 8)  * SPAD_ + k] * vv;
    a3 += sc[(size_t)(rq + 12) * SPAD_ + k] * vv;
  }
  float accs[4] = {a0, a1, a2, a3};
  #pragma unroll
  for (int i = 0; i < 4; ++i) {
    int q = m0 + rq + i * 4;
    if (q < S_)
      o[((size_t)(b * S_ + q) * NHEAD_ + h) * HD_ + d] = accs[i];
  }
}

// ---------------- SwiGLU gate: f = u * silu(g) ----------------
__global__ void silu_gate(const float* __restrict__ uv, float* __restrict__ f) {
  int tid = blockIdx.x * blockDim.x + threadIdx.x;
  if (tid >= MROWS_ * DFF_) return;
  int c = tid % DFF_; int row = tid / DFF_;
  float u = uv[(size_t)row * (2 * DFF_) + c];
  float g = uv[(size_t)row * (2 * DFF_) + DFF_ + c];
  f[(size_t)row * DFF_ + c] = u * (g / (1.f + expf(-g)));
}

// ---------------- host-side launcher ----------------
static inline void launch_gemm(const float* A, const float* Wm, const float* Res, float* C,
                               int M, int K, int N, hipStream_t s) {
  int ntn = N / 16;
  int nquad = (ntn + 3) / 4;
  int tiles = ((M + 15) / 16) * nquad;
  int blocks = (tiles + 7) / 8;                // 8 waves per 256-thread block
  gemm_f32_wmma<<<blocks, 256, 0, s>>>(A, Wm, Res, C, M, K, N);
}

extern "C" void kernel_launch(void* const* d_in, const int* in_sizes, int n_in,
                              void* d_out, int out_size, void* d_ws, size_t ws_size,
                              hipStream_t stream) {
  const float* x    = (const float*)d_in[0];
  const float* frq  = (const float*)d_in[1];
  const float* Wkvd = (const float*)d_in[2];
  const float* Wqd  = (const float*)d_in[3];
  const float* Wkvu = (const float*)d_in[4];
  const float* Wqu  = (const float*)d_in[5];
  const float* kvnw = (const float*)d_in[6];
  const float* qnw  = (const float*)d_in[7];
  const float* Wo   = (const float*)d_in[8];
  const float* lq1  = (const float*)d_in[9];
  const float* lk1  = (const float*)d_in[10];
  const float* lq2  = (const float*)d_in[11];
  const float* lk2  = (const float*)d_in[12];
  const float* hnw  = (const float*)d_in[13];
  const float* n1w  = (const float*)d_in[14];
  const float* n2w  = (const float*)d_in[15];
  const float* Wffi = (const float*)d_in[16];
  const float* Wffo = (const float*)d_in[17];
  float* out = (float*)d_out;
  float* W   = (float*)d_ws;

  const int M = MROWS_;                        // 3074
  // workspace layout (floats), with aliasing of dead regions:
  float* h    = W + 0;                          // 3074*512
  float* ckv  = W + 1573888;                    // 3074*288
  float* qd   = W + 2459200;                    // 3074*384
  float* ckvn = W + 3639616;                    // 3074*256
  float* qdn  = W + 4426560;                    // 3074*384
  float* kv   = W + 5606976;                    // 3074*1024
  float* qf   = W + 8754752;                    // 3074*768
  float* qhB  = W + 11115584;                   // 2*16*1537*48
  float* khB  = W + 13476416;                   // 2*16*1537*48
  float* vT   = W + 15837248;                   // 2*8*1537*64
  float* oB   = W + 17411136;                   // 3074*512
  float* oN   = W + 18985024;                   // 3074*512
  float* x2   = W + 20558912;                   // 3074*512
  float* h2   = h;                              // reuse (h dead after down-proj)
  float* uv   = ckv;                            // reuse ckv..qf region (9.44M floats, fits)
  float* f    = qhB;                            // reuse qh/kh region (4.72M floats)
  // total footprint: 22,132,800 floats = 88.5 MB

  // 1) h = rmsnorm(x, norm1_w)
  rmsnorm_kernel<<<M, 256, 0, stream>>>(x, n1w, h, DM_, DM_, EPS_);
  // 2) ckv = h @ W_kv_down (512 -> 288) ; qd = h @ W_q_down (512 -> 384)
  launch_gemm(h, Wkvd, nullptr, ckv, M, DM_, KVC_ + ROPE_, stream);
  launch_gemm(h, Wqd,  nullptr, qd,  M, DM_, QC_,          stream);
  // 3) rmsnorm latents
  rmsnorm_kernel<<<M, 256, 0, stream>>>(ckv, kvnw, ckvn, KVC_, KVC_ + ROPE_, EPS_);
  rmsnorm_kernel<<<M, 256, 0, stream>>>(qd,  qnw,  qdn,  QC_,  QC_,          EPS_);
  // 4) up-projections
  launch_gemm(ckvn, Wkvu, nullptr, kv, M, KVC_, NHEAD_ * HD_ * 2,      stream);
  launch_gemm(qdn,  Wqu,  nullptr, qf, M, QC_,  NHEAD_ * (HD_ + ROPE_), stream);
  // 5) head-pair split + RoPE ; V extraction
  {
    int tot = B_ * S_ * H2_;
    build_qk<<<(tot + 255) / 256, 256, 0, stream>>>(qf, kv, ckv, frq, qhB, khB);
    int tv = B_ * S_ * NHEAD_ * HD_;
    extract_v<<<(tv + 255) / 256, 256, 0, stream>>>(kv, vT);
  }
  // 6) differential attention
  {
    size_t smem = (size_t)(2 * 16 * SPAD_ + 32) * sizeof(float);  // ~193 KB (CDNA5: 320KB/WGP)
    dim3 grid(NKT_, B_ * NHEAD_);
    attention_kernel<<<grid, 256, smem, stream>>>(qhB, khB, vT, lq1, lk1, lq2, lk2, oB);
  }
  // 7) per-head rmsnorm (eps 1e-5) then O-projection with residual x
  rmsnorm_kernel<<<M * NHEAD_, 64, 0, stream>>>(oB, hnw, oN, HD_, HD_, 1e-5f);
  launch_gemm(oN, Wo, x, x2, M, DM_, DM_, stream);
  // 8) FFN: h2 = rmsnorm(x2); uv = h2 @ W_ff_in; f = u*silu(g); out = f @ W_ff_out + x2
  rmsnorm_kernel<<<M, 256, 0, stream>>>(x2, n2w, h2, DM_, DM_, EPS_);
  launch_gemm(h2, Wffi, nullptr, uv, M, DM_, 2 * DFF_, stream);
  {
    int tot = M * DFF_;
    silu_gate<<<(tot + 255) / 256, 256, 0, stream>>>(uv, f);
  }
  launch_gemm(f, Wffo, x2, out, M, DFF_, DM_, stream);
}